// HGTLayer_71545565217540
// MI455X (gfx1250) — compile-verified
//
#include <hip/hip_runtime.h>

typedef __attribute__((ext_vector_type(2))) float v2f;
typedef __attribute__((ext_vector_type(8))) float v8f;

#define LDSS 132  // 128 + 4 padding to avoid LDS bank conflicts

__device__ __forceinline__ v8f wmma_f32_k4(v2f a, v2f b, v8f c) {
  // D = A(16x4) * B(4x16) + C(16x16), fp32
  return __builtin_amdgcn_wmma_f32_16x16x4_f32(false, a, false, b, (short)0, c,
                                               false, false);
}

// ---------------- GEMM: Y[N,128] = X[N,128] * W[128,128] + bias ----------------
__global__ __launch_bounds__(256) void hgt_proj_gemm(
    const float* __restrict__ X, const float* __restrict__ W,
    const float* __restrict__ bias, float* __restrict__ Y, int N, float xscale) {
  __shared__ float lds[16 * LDSS];
  const int tid = threadIdx.x;
  const int wave = tid >> 5, lane = tid & 31;
  const int half = lane >> 4, l16 = lane & 15;
  const long node0 = (long)blockIdx.x * 16;

  for (int i = tid; i < 16 * 128; i += 256) {
    int r = i >> 7, c = i & 127;
    long row = node0 + r;
    if (row >= N) row = N - 1;
    lds[r * LDSS + c] = X[row * 128 + c] * xscale;
  }
  __syncthreads();

  const int col0 = wave * 16;
  v8f acc = {};
#pragma unroll 8
  for (int k = 0; k < 128; k += 4) {
    v2f a, b;
    a.x = lds[l16 * LDSS + k + half * 2 + 0];
    a.y = lds[l16 * LDSS + k + half * 2 + 1];
    b.x = W[(k + half * 2 + 0) * 128 + col0 + l16];
    b.y = W[(k + half * 2 + 1) * 128 + col0 + l16];
    acc = wmma_f32_k4(a, b, acc);
  }
  const float bv = bias[col0 + l16];
  float* Yp = Y + (node0 + half * 8) * 128 + col0 + l16;
  if (node0 + 16 <= (long)N) {  // uniform fast path: full tile
#pragma unroll
    for (int v = 0; v < 8; ++v) Yp[v * 128] = acc[v] + bv;
  } else {
#pragma unroll
    for (int v = 0; v < 8; ++v) {
      if (node0 + v + half * 8 < (long)N) Yp[v * 128] = acc[v] + bv;
    }
  }
}

// ------------- Output GEMM + sigmoid-skip residual mix -------------
__global__ __launch_bounds__(256) void hgt_out_gemm(
    const float* __restrict__ AGG, const float* __restrict__ W,
    const float* __restrict__ bias, const float* __restrict__ Hin,
    const float* __restrict__ skip, float* __restrict__ Y, int N, float xscale) {
  __shared__ float lds[16 * LDSS];
  const int tid = threadIdx.x;
  const int wave = tid >> 5, lane = tid & 31;
  const int half = lane >> 4, l16 = lane & 15;
  const long node0 = (long)blockIdx.x * 16;
  const float alpha = 1.0f / (1.0f + __expf(-skip[0]));

  for (int i = tid; i < 16 * 128; i += 256) {
    int r = i >> 7, c = i & 127;
    long row = node0 + r;
    if (row >= N) row = N - 1;
    lds[r * LDSS + c] = AGG[row * 128 + c] * xscale;
  }
  __syncthreads();

  const int col0 = wave * 16;
  v8f acc = {};
#pragma unroll 8
  for (int k = 0; k < 128; k += 4) {
    v2f a, b;
    a.x = lds[l16 * LDSS + k + half * 2 + 0];
    a.y = lds[l16 * LDSS + k + half * 2 + 1];
    b.x = W[(k + half * 2 + 0) * 128 + col0 + l16];
    b.y = W[(k + half * 2 + 1) * 128 + col0 + l16];
    acc = wmma_f32_k4(a, b, acc);
  }
  const float bv = bias[col0 + l16];
  const long base = (node0 + half * 8) * 128 + col0 + l16;
  if (node0 + 16 <= (long)N) {  // uniform fast path: full tile
#pragma unroll
    for (int v = 0; v < 8; ++v) {
      float t = acc[v] + bv;
      Y[base + v * 128] = t * alpha + Hin[base + v * 128] * (1.0f - alpha);
    }
  } else {
#pragma unroll
    for (int v = 0; v < 8; ++v) {
      if (node0 + v + half * 8 < (long)N) {
        float t = acc[v] + bv;
        Y[base + v * 128] = t * alpha + Hin[base + v * 128] * (1.0f - alpha);
      }
    }
  }
}

// ------- Per-head 16x16 relation transform: KK = K @ rel_att, VV = V @ rel_msg -------
__global__ __launch_bounds__(256) void hgt_rel_transform(
    const float* __restrict__ Kin, const float* __restrict__ Vin,
    const float* __restrict__ RelAtt, const float* __restrict__ RelMsg,
    float* __restrict__ KK, float* __restrict__ VV, int N) {
  const float* S = blockIdx.y ? Vin : Kin;
  const float* R = blockIdx.y ? RelMsg : RelAtt;
  float* O = blockIdx.y ? VV : KK;
  const int tid = threadIdx.x;
  const int h = tid >> 5;  // one wave per head (8 waves)
  const int lane = tid & 31;
  const int half = lane >> 4, l16 = lane & 15;
  const long node0 = (long)blockIdx.x * 16;

  long arow = node0 + l16;
  if (arow >= N) arow = N - 1;

  v8f acc = {};
#pragma unroll
  for (int k = 0; k < 16; k += 4) {
    v2f a, b;
    a.x = S[arow * 128 + h * 16 + k + half * 2 + 0];
    a.y = S[arow * 128 + h * 16 + k + half * 2 + 1];
    b.x = R[h * 256 + (k + half * 2 + 0) * 16 + l16];
    b.y = R[h * 256 + (k + half * 2 + 1) * 16 + l16];
    acc = wmma_f32_k4(a, b, acc);
  }
  float* Op = O + (node0 + half * 8) * 128 + h * 16 + l16;
  if (node0 + 16 <= (long)N) {  // uniform fast path
#pragma unroll
    for (int v = 0; v < 8; ++v) Op[v * 128] = acc[v];
  } else {
#pragma unroll
    for (int v = 0; v < 8; ++v) {
      if (node0 + v + half * 8 < (long)N) Op[v * 128] = acc[v];
    }
  }
}

// ---------------- Edge-softmax helpers ----------------
__device__ __forceinline__ unsigned f2mono(float f) {
  unsigned b = __float_as_uint(f);
  return (b & 0x80000000u) ? ~b : (b | 0x80000000u);
}
__device__ __forceinline__ float mono2f(unsigned u) {
  return __uint_as_float((u & 0x80000000u) ? (u & 0x7FFFFFFFu) : ~u);
}

__global__ void hgt_fill_u32(unsigned* __restrict__ p, unsigned val, long n) {
  long i = (long)blockIdx.x * blockDim.x + threadIdx.x;
  if (i < n) p[i] = val;
}

__global__ void hgt_edge_score(const float* __restrict__ q,
                               const float* __restrict__ kk,
                               const int* __restrict__ src,
                               const int* __restrict__ dst,
                               const float* __restrict__ pri,
                               float* __restrict__ score,
                               unsigned* __restrict__ m, int E) {
  int i = blockIdx.x * blockDim.x + threadIdx.x;
  if (i >= E * 8) return;
  int e = i >> 3, h = i & 7;
  int s = src[e], d = dst[e];
  const float4* qp = (const float4*)(q + (long)d * 128 + h * 16);
  const float4* kp = (const float4*)(kk + (long)s * 128 + h * 16);
  float acc = 0.0f;
#pragma unroll
  for (int c = 0; c < 4; ++c) {
    float4 qv = qp[c], kv = kp[c];
    acc += qv.x * kv.x + qv.y * kv.y + qv.z * kv.z + qv.w * kv.w;
  }
  float sc = acc * pri[h] * 0.25f;  // / sqrt(d_k), d_k = 16
  score[i] = sc;
  atomicMax(&m[(long)d * 8 + h], f2mono(sc));
}

__global__ void hgt_edge_exp(const int* __restrict__ dst,
                             float* __restrict__ score,
                             const unsigned* __restrict__ m,
                             float* __restrict__ ssum, int E) {
  int i = blockIdx.x * blockDim.x + threadIdx.x;
  if (i >= E * 8) return;
  int e = i >> 3, h = i & 7;
  int d = dst[e];
  float mm = mono2f(m[(long)d * 8 + h]);
  float ev = __expf(score[i] - mm);
  score[i] = ev;
  atomicAdd(&ssum[(long)d * 8 + h], ev);
}

__global__ void hgt_edge_scatter(const int* __restrict__ src,
                                 const int* __restrict__ dst,
                                 const float* __restrict__ vv,
                                 const float* __restrict__ score,
                                 const float* __restrict__ ssum,
                                 float* __restrict__ agg, int E) {
  int i = blockIdx.x * blockDim.x + threadIdx.x;
  if (i >= E * 8) return;
  int e = i >> 3, h = i & 7;
  int s = src[e], d = dst[e];
  float a = score[i] / ssum[(long)d * 8 + h];
  const float4* vp = (const float4*)(vv + (long)s * 128 + h * 16);
  float* ap = agg + (long)d * 128 + h * 16;
#pragma unroll
  for (int c = 0; c < 4; ++c) {
    float4 vval = vp[c];
    atomicAdd(&ap[c * 4 + 0], vval.x * a);
    atomicAdd(&ap[c * 4 + 1], vval.y * a);
    atomicAdd(&ap[c * 4 + 2], vval.z * a);
    atomicAdd(&ap[c * 4 + 3], vval.w * a);
  }
}

// ---------------- Host-side orchestration ----------------
extern "C" void kernel_launch(void* const* d_in, const int* in_sizes, int n_in,
                              void* d_out, int out_size, void* d_ws,
                              size_t ws_size, hipStream_t stream) {
  const float* h0 = (const float*)d_in[0];
  const float* h1 = (const float*)d_in[1];
  const float* kw = (const float*)d_in[2];
  const float* kb = (const float*)d_in[3];
  const float* qw = (const float*)d_in[4];
  const float* qb = (const float*)d_in[5];
  const float* vw = (const float*)d_in[6];
  const float* vb = (const float*)d_in[7];
  const float* aw = (const float*)d_in[8];
  const float* ab = (const float*)d_in[9];
  const float* rel_att = (const float*)d_in[10];
  const float* rel_msg = (const float*)d_in[11];
  const float* rel_pri = (const float*)d_in[12];
  const float* skip = (const float*)d_in[13];
  const int* srcs[3] = {(const int*)d_in[14], (const int*)d_in[16],
                        (const int*)d_in[18]};
  const int* dsts[3] = {(const int*)d_in[15], (const int*)d_in[17],
                        (const int*)d_in[19]};

  const int N0 = in_sizes[0] / 128;
  const int N1 = in_sizes[1] / 128;
  const int E = in_sizes[14];
  const long NF0 = (long)N0 * 128, NF1 = (long)N1 * 128;
  const long NFmax = NF0 > NF1 ? NF0 : NF1;
  const long NHmax = (long)(N0 > N1 ? N0 : N1) * 8;

  float* p = (float*)d_ws;
  float* k0 = p;   p += NF0;
  float* q0 = p;   p += NF0;
  float* v0 = p;   p += NF0;
  float* k1 = p;   p += NF1;
  float* q1 = p;   p += NF1;
  float* v1 = p;   p += NF1;
  float* kk = p;   p += NFmax;
  float* vv = p;   p += NFmax;
  float* agg0 = p; p += NF0;
  float* agg1 = p; p += NF1;
  unsigned* mbuf = (unsigned*)p; p += NHmax;
  float* sbuf = p; p += NHmax;
  float* score = p;  // E*8 floats

  dim3 blk(256);
  const int tiles0 = (N0 + 15) / 16, tiles1 = (N1 + 15) / 16;

  // k/q/v projections for both node types
  hgt_proj_gemm<<<tiles0, blk, 0, stream>>>(h0, kw, kb, k0, N0, 1.0f);
  hgt_proj_gemm<<<tiles0, blk, 0, stream>>>(h0, qw, qb, q0, N0, 1.0f);
  hgt_proj_gemm<<<tiles0, blk, 0, stream>>>(h0, vw, vb, v0, N0, 1.0f);
  hgt_proj_gemm<<<tiles1, blk, 0, stream>>>(h1, kw + 128 * 128, kb + 128, k1, N1, 1.0f);
  hgt_proj_gemm<<<tiles1, blk, 0, stream>>>(h1, qw + 128 * 128, qb + 128, q1, N1, 1.0f);
  hgt_proj_gemm<<<tiles1, blk, 0, stream>>>(h1, vw + 128 * 128, vb + 128, v1, N1, 1.0f);

  // zero aggregation buffers (agg0/agg1 contiguous)
  {
    long n = NF0 + NF1;
    hgt_fill_u32<<<(int)((n + 255) / 256), blk, 0, stream>>>((unsigned*)agg0, 0u, n);
  }

  // relations: (r, k/v src, q dst, Nsrc, Ndst, agg)
  const float* rk[3] = {k1, k0, k0};
  const float* rv[3] = {v1, v0, v0};
  const float* rq[3] = {q0, q1, q0};
  const int rNs[3] = {N1, N0, N0};
  float* ragg[3] = {agg0, agg1, agg0};

  const int egrid = (E * 8 + 255) / 256;
  for (int r = 0; r < 3; ++r) {
    const int tilesS = (rNs[r] + 15) / 16;
    hgt_rel_transform<<<dim3(tilesS, 2), blk, 0, stream>>>(
        rk[r], rv[r], rel_att + (long)r * 8 * 256, rel_msg + (long)r * 8 * 256,
        kk, vv, rNs[r]);
    // reset m (monotone-encoded -inf == 0) and s (0.0f); contiguous buffers
    long n = 2 * NHmax;
    hgt_fill_u32<<<(int)((n + 255) / 256), blk, 0, stream>>>(mbuf, 0u, n);
    hgt_edge_score<<<egrid, blk, 0, stream>>>(rq[r], kk, srcs[r], dsts[r],
                                              rel_pri + r * 8, score, mbuf, E);
    hgt_edge_exp<<<egrid, blk, 0, stream>>>(dsts[r], score, mbuf, sbuf, E);
    hgt_edge_scatter<<<egrid, blk, 0, stream>>>(srcs[r], dsts[r], vv, score,
                                                sbuf, ragg[r], E);
  }

  // output GEMM + skip mix; type 0 averages two relations -> scale 0.5
  float* out = (float*)d_out;
  hgt_out_gemm<<<tiles0, blk, 0, stream>>>(agg0, aw, ab, h0, skip, out, N0, 0.5f);
  hgt_out_gemm<<<tiles1, blk, 0, stream>>>(agg1, aw + 128 * 128, ab + 128, h1,
                                           skip + 1, out + NF0, N1, 1.0f);
}